// Histogram1D_10582799417522
// MI455X (gfx1250) — compile-verified
//
#include <hip/hip_runtime.h>
#include <hip/hip_bf16.h>

#define NBINS 128
#define NDIM  6

typedef __attribute__((ext_vector_type(2))) float v2f;
typedef __attribute__((ext_vector_type(8))) float v8f;

// ---------------------------------------------------------------------------
// Kernel 1: per-workgroup partial KDE histograms.
// Each point contributes only to bins c-7 .. c+8 (tails < 1e-12), so we do
// 17 erf evaluations per point instead of 129. Contributions are scattered
// into a 128-entry LDS f32 histogram with ds_add_f32 atomics; the block's
// partial histogram is written to workspace for the WMMA reduction kernel.
// ---------------------------------------------------------------------------
__global__ void kde_hist_partial(const float* __restrict__ x,
                                 const float* __restrict__ edges,
                                 float* __restrict__ partial,
                                 int n) {
    __shared__ float lhist[NBINS];
    for (int j = threadIdx.x; j < NBINS; j += blockDim.x) lhist[j] = 0.0f;
    __syncthreads();

    const float e0      = edges[0];
    const float res     = edges[1] - e0;
    const float inv_res = 1.0f / res;                 // bandwidth == res
    const float INV_SQRT2 = 0.7071067811865476f;

    const int stride = gridDim.x * blockDim.x;
    for (int i = blockIdx.x * blockDim.x + threadIdx.x; i < n; i += stride) {
        const float xv = x[(size_t)i * NDIM];          // column AXIS=0, stride 6
        const float t  = (xv - e0) * inv_res;          // position in bin units
        const int   c  = (int)floorf(t);

        int jlo = c - 7;  if (jlo < 0)     jlo = 0;  if (jlo > NBINS) jlo = NBINS;
        int jhi = c + 9;  if (jhi > NBINS) jhi = NBINS;
        if (jhi < jlo) jhi = jlo;

        // z at edge j is (j - t)/sqrt(2); consecutive edges differ by 1/sqrt(2)
        float u    = ((float)jlo - t) * INV_SQRT2;
        float prev = 0.5f * (1.0f + erff(u));
        for (int j = jlo; j < jhi; ++j) {
            u += INV_SQRT2;
            const float cur = 0.5f * (1.0f + erff(u));
            atomicAdd(&lhist[j], cur - prev);          // ds_add_f32 (no return)
            prev = cur;
        }
    }

    __syncthreads();
    for (int j = threadIdx.x; j < NBINS; j += blockDim.x)
        partial[(size_t)blockIdx.x * NBINS + j] = lhist[j];
}

// ---------------------------------------------------------------------------
// Kernel 2: reduce partial histograms with V_WMMA_F32_16X16X4_F32.
// s[j] = sum_b partial[b][j]  ==  ones(16x4) x partial_chunk(4x16) accumulated.
// A = all-ones, so D[m,n] = sum_k B[k,n] for every m; results are exact f32.
// One block, 8 waves; wave w reduces the 16-bin tile [16w, 16w+16).
// Then a single LDS pass computes the total and the final normalization
// out[j] = s[j] / (sum(s) * res)   (the 1/N of the mean cancels).
// ---------------------------------------------------------------------------
__global__ void kde_hist_reduce(const float* __restrict__ partial,
                                const float* __restrict__ edges,
                                float* __restrict__ out,
                                int nblk) {
    __shared__ float s[NBINS + 1];

    const int tid  = threadIdx.x;
    const int wave = tid >> 5;          // 0..7 -> bin tile
    const int lane = tid & 31;
    const int nn   = lane & 15;         // N index within tile (lane striping)
    const int kh   = lane >> 4;         // K half: lanes 0-15 hold K={0,1}, 16-31 hold K={2,3}

    v2f a; a.x = 1.0f; a.y = 1.0f;      // A-matrix: all ones (any layout works)
    v8f acc = {};

    const float* colbase = partial + wave * 16 + nn;
    for (int b = 0; b < nblk; b += 4) {
        const int r0 = b + 2 * kh;
        v2f bb;
        bb.x = colbase[(size_t)(r0    ) * NBINS];
        bb.y = colbase[(size_t)(r0 + 1) * NBINS];
        acc = __builtin_amdgcn_wmma_f32_16x16x4_f32(
            /*neg_a=*/false, a, /*neg_b=*/false, bb,
            /*c_mod=*/(short)0, acc, /*reuse_a=*/false, /*reuse_b=*/false);
    }

    // D row M=0 lives in acc[0], lanes 0-15 (documented C/D layout); rows are
    // identical anyway since A is all-ones.
    if (lane < 16) s[wave * 16 + nn] = acc[0];
    __syncthreads();

    if (tid == 0) {
        float tot = 0.0f;
        for (int j = 0; j < NBINS; ++j) tot += s[j];
        s[NBINS] = tot;
    }
    __syncthreads();

    const float res = edges[1] - edges[0];
    if (tid < NBINS) out[tid] = s[tid] / (s[NBINS] * res);
}

// ---------------------------------------------------------------------------
extern "C" void kernel_launch(void* const* d_in, const int* in_sizes, int n_in,
                              void* d_out, int out_size, void* d_ws, size_t ws_size,
                              hipStream_t stream) {
    const float* x     = (const float*)d_in[0];
    const float* edges = (const float*)d_in[1];
    float*       out   = (float*)d_out;
    float*       ws    = (float*)d_ws;

    const int npoints = in_sizes[0] / NDIM;

    // Partial histograms: one 128-float row per block. Size the grid from the
    // workspace (multiple of 4 rows for the 16x16x4 WMMA reduction chunks).
    int nblk = (int)(ws_size / (NBINS * sizeof(float)));
    if (nblk > 1024) nblk = 1024;
    nblk &= ~3;
    if (nblk < 4) nblk = 4;

    kde_hist_partial<<<nblk, 256, 0, stream>>>(x, edges, ws, npoints);
    kde_hist_reduce<<<1, 256, 0, stream>>>(ws, edges, out, nblk);
}